// HeteroGCN_40939628265721
// MI455X (gfx1250) — compile-verified
//
#include <hip/hip_runtime.h>

#define NN 40000
#define NE 640000
#define SLOPE 0.2f

typedef __attribute__((ext_vector_type(2))) float v2f;
typedef __attribute__((ext_vector_type(8))) float v8f;

// ---------------- dense GEMM: H[N,F] = X[N,K] @ W[K,F]  (f32 WMMA 16x16x4) ----
// One wave computes a full 16-row x F output strip: TF = F/16 accumulators live
// in registers; the A fragment is loaded once per k-step and reused across all
// column tiles (X read exactly once from HBM; W stays L2/WGP$ resident).
template <int TF>
__global__ void gemm_wmma_strip(const float* __restrict__ X, const float* __restrict__ W,
                                float* __restrict__ H, int N, int K) {
  const int F    = TF * 16;
  const int wid  = (blockIdx.x * blockDim.x + threadIdx.x) >> 5;   // strip id
  const int lane = threadIdx.x & 31;
  if (wid * 16 >= N) return;                  // wave-uniform guard
  const int m   = wid * 16 + (lane & 15);     // A row for this lane
  const int nl  = lane & 15;                  // local column within a tile
  const int kh  = (lane >> 4) << 1;           // lanes 0-15 -> K{0,1}; 16-31 -> K{2,3}

  v8f acc[TF];
#pragma unroll
  for (int t = 0; t < TF; ++t) acc[t] = (v8f){};

  for (int k0 = 0; k0 < K; k0 += 4) {
    v2f a;
    a.x = X[m * K + k0 + kh];
    a.y = X[m * K + k0 + kh + 1];
    const float* Wrow0 = W + (k0 + kh) * F;
    const float* Wrow1 = W + (k0 + kh + 1) * F;
#pragma unroll
    for (int t = 0; t < TF; ++t) {
      v2f b;
      b.x = Wrow0[t * 16 + nl];
      b.y = Wrow1[t * 16 + nl];
      acc[t] = __builtin_amdgcn_wmma_f32_16x16x4_f32(false, a, false, b, (short)0,
                                                     acc[t], false, false);
    }
  }
  // C/D layout: VGPR v -> row M=v (lanes 0-15) or M=v+8 (lanes 16-31)
  const int rbase = wid * 16 + ((lane >> 4) << 3);
#pragma unroll
  for (int t = 0; t < TF; ++t)
#pragma unroll
    for (int v = 0; v < 8; ++v)
      H[(rbase + v) * F + t * 16 + nl] = acc[t][v];
}

// ---------------- small utility kernels ------------------------------------
__global__ void fill_zero_f(float* p, int n) {
  int i = blockIdx.x * blockDim.x + threadIdx.x;
  if (i < n) p[i] = 0.0f;
}
__global__ void fill_zero_i(int* p, int n) {
  int i = blockIdx.x * blockDim.x + threadIdx.x;
  if (i < n) p[i] = 0;
}
__global__ void degree_kernel(const int* __restrict__ dst, int* __restrict__ deg, int nE) {
  int e = blockIdx.x * blockDim.x + threadIdx.x;
  if (e < nE) atomicAdd(&deg[dst[e]], 1);
}
__global__ void dinv_kernel(const int* __restrict__ deg, float* __restrict__ dinv, int n) {
  int i = blockIdx.x * blockDim.x + threadIdx.x;
  if (i < n) dinv[i] = rsqrtf((float)(deg[i] + 1));   // +1 self loop
}

// ---------------- GCN: scatter + finalize ----------------------------------
__global__ void gcn_agg(const float* __restrict__ h, const int* __restrict__ src,
                        const int* __restrict__ dst, const float* __restrict__ dinv,
                        float* __restrict__ out, int F, int nE) {
  const int wid  = (blockIdx.x * blockDim.x + threadIdx.x) >> 5;
  const int lane = threadIdx.x & 31;
  if (wid >= nE) return;
  const int s = src[wid], d = dst[wid];
  const float w = dinv[s] * dinv[d];
  for (int f = lane; f < F; f += 32)
    atomicAdd(&out[d * F + f], h[s * F + f] * w);
}
__global__ void gcn_finalize(float* __restrict__ out, const float* __restrict__ h,
                             const float* __restrict__ dinv, const float* __restrict__ bias,
                             int F, int n) {
  int idx = blockIdx.x * blockDim.x + threadIdx.x;
  if (idx >= n * F) return;
  int i = idx / F, f = idx % F;
  float v = out[idx] + h[idx] * dinv[i] * dinv[i] + bias[f];
  out[idx] = fmaxf(v, 0.0f);                         // ReLU (both GCN layers)
}

// ---------------- GAT -------------------------------------------------------
__device__ inline void atomicMaxFloat(float* addr, float val) {
  if (val >= 0.0f) atomicMax((int*)addr, __float_as_int(val));
  else             atomicMin((unsigned int*)addr, __float_as_uint(val));
}
__device__ inline float lrelu(float x) { return x > 0.0f ? x : SLOPE * x; }

// per-node dots: as[i]=h[i].a_src, ad[i]=h[i].a_dst; seed mx with self-loop logit
__global__ void gat_dots(const float* __restrict__ h, const float* __restrict__ a_src,
                         const float* __restrict__ a_dst, float* __restrict__ as,
                         float* __restrict__ ad, float* __restrict__ selfl,
                         float* __restrict__ mx, int F, int n) {
  const int node = (blockIdx.x * blockDim.x + threadIdx.x) >> 5;
  const int lane = threadIdx.x & 31;
  if (node >= n) return;
  float s1 = 0.0f, s2 = 0.0f;
  for (int f = lane; f < F; f += 32) {
    float v = h[node * F + f];
    s1 += v * a_src[f];
    s2 += v * a_dst[f];
  }
#pragma unroll
  for (int off = 16; off > 0; off >>= 1) {
    s1 += __shfl_down(s1, off, 32);
    s2 += __shfl_down(s2, off, 32);
  }
  if (lane == 0) {
    as[node] = s1;
    ad[node] = s2;
    float l = lrelu(s1 + s2);
    selfl[node] = l;
    mx[node] = l;
  }
}
__global__ void gat_logit_max(const int* __restrict__ src, const int* __restrict__ dst,
                              const float* __restrict__ as, const float* __restrict__ ad,
                              float* __restrict__ elog, float* __restrict__ mx, int nE) {
  int e = blockIdx.x * blockDim.x + threadIdx.x;
  if (e >= nE) return;
  float l = lrelu(as[src[e]] + ad[dst[e]]);
  elog[e] = l;
  atomicMaxFloat(&mx[dst[e]], l);
}
__global__ void gat_node_exp(const float* __restrict__ selfl, const float* __restrict__ mx,
                             float* __restrict__ esf, float* __restrict__ z, int n) {
  int i = blockIdx.x * blockDim.x + threadIdx.x;
  if (i >= n) return;
  float e = expf(selfl[i] - mx[i]);
  esf[i] = e;
  z[i] = e;
}
__global__ void gat_edge_exp(const int* __restrict__ dst, const float* __restrict__ mx,
                             float* __restrict__ elog, float* __restrict__ z, int nE) {
  int e = blockIdx.x * blockDim.x + threadIdx.x;
  if (e >= nE) return;
  int d = dst[e];
  float v = expf(elog[e] - mx[d]);
  elog[e] = v;
  atomicAdd(&z[d], v);
}
__global__ void recip_kernel(float* __restrict__ z, int n) {
  int i = blockIdx.x * blockDim.x + threadIdx.x;
  if (i < n) z[i] = 1.0f / z[i];
}
// out = self-loop term + bias (full overwrite -> deterministic init)
__global__ void gat_init_out(const float* __restrict__ h, const float* __restrict__ esf,
                             const float* __restrict__ zinv, const float* __restrict__ bias,
                             float* __restrict__ out, int F, int n) {
  int idx = blockIdx.x * blockDim.x + threadIdx.x;
  if (idx >= n * F) return;
  int i = idx / F, f = idx % F;
  out[idx] = h[idx] * esf[i] * zinv[i] + bias[f];
}
__global__ void gat_agg(const float* __restrict__ h, const int* __restrict__ src,
                        const int* __restrict__ dst, const float* __restrict__ ee,
                        const float* __restrict__ zinv, float* __restrict__ out,
                        int F, int nE) {
  const int wid  = (blockIdx.x * blockDim.x + threadIdx.x) >> 5;
  const int lane = threadIdx.x & 31;
  if (wid >= nE) return;
  const int s = src[wid], d = dst[wid];
  const float w = ee[wid] * zinv[d];
  for (int f = lane; f < F; f += 32)
    atomicAdd(&out[d * F + f], h[s * F + f] * w);
}

// ---------------- host-side orchestration ----------------------------------
static inline int cdiv(int a, int b) { return (a + b - 1) / b; }

static void run_gemm(const float* X, const float* W, float* H, int K, int F,
                     hipStream_t stream) {
  int strips = NN / 16;                        // one wave per 16-row strip
  int blocks = cdiv(strips * 32, 256);
  if (F == 128)
    gemm_wmma_strip<8><<<blocks, 256, 0, stream>>>(X, W, H, NN, K);
  else
    gemm_wmma_strip<4><<<blocks, 256, 0, stream>>>(X, W, H, NN, K);
}

static void run_gcn(const float* xin, const float* W, const float* bias, int K, int F,
                    const int* src, const int* dst, const float* dinv,
                    float* hbuf, float* outbuf, hipStream_t stream) {
  run_gemm(xin, W, hbuf, K, F, stream);
  fill_zero_f<<<cdiv(NN * F, 256), 256, 0, stream>>>(outbuf, NN * F);
  gcn_agg<<<NE / 8, 256, 0, stream>>>(hbuf, src, dst, dinv, outbuf, F, NE);
  gcn_finalize<<<cdiv(NN * F, 256), 256, 0, stream>>>(outbuf, hbuf, dinv, bias, F, NN);
}

static void run_gat(const float* xin, const float* W, const float* avs, const float* avd,
                    const float* bias, int K, int F, const int* src, const int* dst,
                    float* hbuf, float* elog, float* asr, float* ads, float* slg,
                    float* mx, float* zz, float* esf, float* outbuf, hipStream_t stream) {
  run_gemm(xin, W, hbuf, K, F, stream);
  gat_dots<<<NN / 8, 256, 0, stream>>>(hbuf, avs, avd, asr, ads, slg, mx, F, NN);
  gat_logit_max<<<cdiv(NE, 256), 256, 0, stream>>>(src, dst, asr, ads, elog, mx, NE);
  gat_node_exp<<<cdiv(NN, 256), 256, 0, stream>>>(slg, mx, esf, zz, NN);
  gat_edge_exp<<<cdiv(NE, 256), 256, 0, stream>>>(dst, mx, elog, zz, NE);
  recip_kernel<<<cdiv(NN, 256), 256, 0, stream>>>(zz, NN);
  gat_init_out<<<cdiv(NN * F, 256), 256, 0, stream>>>(hbuf, esf, zz, bias, outbuf, F, NN);
  gat_agg<<<NE / 8, 256, 0, stream>>>(hbuf, src, dst, elog, zz, outbuf, F, NE);
}

extern "C" void kernel_launch(void* const* d_in, const int* in_sizes, int n_in,
                              void* d_out, int out_size, void* d_ws, size_t ws_size,
                              hipStream_t stream) {
  const float* x   = (const float*)d_in[0];
  const int*   ei  = (const int*)d_in[1];
  const int* src = ei;          // edge_index[0]
  const int* dst = ei + NE;     // edge_index[1]
  const float* W1  = (const float*)d_in[2];
  const float* b1  = (const float*)d_in[3];
  const float* Wg1 = (const float*)d_in[4];
  const float* a1s = (const float*)d_in[5];
  const float* a1d = (const float*)d_in[6];
  const float* bg1 = (const float*)d_in[7];
  const float* W2  = (const float*)d_in[8];
  const float* b2  = (const float*)d_in[9];
  const float* Wg2 = (const float*)d_in[10];
  const float* a2s = (const float*)d_in[11];
  const float* a2d = (const float*)d_in[12];
  const float* bg2 = (const float*)d_in[13];

  float* enc = (float*)d_out;          // [40000,128]
  float* dec = enc + NN * 128;         // [40000,64]

  // workspace layout (floats)
  float* ws   = (float*)d_ws;
  float* hA   = ws;                    // GEMM out (h1 / h3)
  float* hB   = hA + NN * 128;         // GCN out  (g1 / g2)
  float* hC   = hB + NN * 128;         // GAT feat (h2 / h4)
  float* elog = hC + NN * 128;         // per-edge logits / exp
  int*   deg  = (int*)(elog + NE);
  float* dinv = (float*)(deg + NN);
  float* asr  = dinv + NN;
  float* ads  = asr + NN;
  float* slg  = ads + NN;
  float* mx   = slg + NN;
  float* zz   = mx + NN;
  float* esf  = zz + NN;

  // degrees -> dinv (recomputed every call; deterministic)
  fill_zero_i<<<cdiv(NN, 256), 256, 0, stream>>>(deg, NN);
  degree_kernel<<<cdiv(NE, 256), 256, 0, stream>>>(dst, deg, NE);
  dinv_kernel<<<cdiv(NN, 256), 256, 0, stream>>>(deg, dinv, NN);

  // layer 1: GCN(128->128)+ReLU, then GAT(128->128) -> encoded
  run_gcn(x, W1, b1, 128, 128, src, dst, dinv, hA, hB, stream);
  run_gat(hB, Wg1, a1s, a1d, bg1, 128, 128, src, dst,
          hC, elog, asr, ads, slg, mx, zz, esf, enc, stream);

  // layer 2: GCN(128->64)+ReLU, then GAT(64->64) -> decoded
  run_gcn(enc, W2, b2, 128, 64, src, dst, dinv, hA, hB, stream);
  run_gat(hB, Wg2, a2s, a2d, bg2, 64, 64, src, dst,
          hC, elog, asr, ads, slg, mx, zz, esf, dec, stream);
}